// IndRNN_onlyrecurrent_19250043420969
// MI455X (gfx1250) — compile-verified
//
#include <hip/hip_runtime.h>

// IndRNN diagonal recurrence: h_t = relu(x_t + w ⊙ h_{t-1}), streamed.
// Bandwidth-bound: 512 MB total traffic, ~22 us floor at 23.3 TB/s.
//
// 3-stage CDNA5 pipeline per wave:
//   global_prefetch_b8 (dist 56, HBM->L2, counter-free)
//   global_load_async_to_lds_b128 (depth 24 ring, ASYNCcnt-tracked, L2->LDS)
//   ds_load_b128 -> v_fma/v_max -> nontemporal global_store_b128

typedef float v4f __attribute__((ext_vector_type(4)));

#define T_STEPS 1024
#define H_DIM   1024
#define B_DIM   64
#define BH      (B_DIM * H_DIM)   // 65536 channels
#define S4      (BH / 4)          // float4 columns per timestep = 16384
#define BLK     64                // threads per block (2 waves)
#define RING    32                // LDS ring slots (power of 2)
#define DEPTH   24                // async loads in flight per wave (<= 63)
#define PF_DIST 56                // prefetch distance in timesteps (> DEPTH)

#define ASYNC_LOAD_B128(ldsaddr, gptr)                                   \
    asm volatile("global_load_async_to_lds_b128 %0, %1, off"             \
                 :: "v"(ldsaddr), "v"(gptr) : "memory")

__global__ __launch_bounds__(BLK) void indrnn_async_kernel(
    const float* __restrict__ x,    // [T, B, H]
    const float* __restrict__ h0,   // [B, H]
    const float* __restrict__ w,    // [H]
    float* __restrict__ out)        // [T, B, H]
{
    __shared__ v4f ring[RING * BLK];   // 32 KB: [slot][lane] of float4

    const int tid  = threadIdx.x;
    const int c4   = blockIdx.x * BLK + tid;   // float4 column in B*H
    const int flat = c4 * 4;
    const int hidx = flat & (H_DIM - 1);

    v4f wv = *reinterpret_cast<const v4f*>(w + hidx);
    v4f hv = *reinterpret_cast<const v4f*>(h0 + flat);

    const v4f* xp = reinterpret_cast<const v4f*>(x) + c4;
    v4f*       op = reinterpret_cast<v4f*>(out) + c4;

    // This lane's LDS byte address for ring slot 0 (generic shared pointer:
    // low 32 bits == LDS byte offset). Slot stride = BLK lanes * 16 B.
    const unsigned lds_lane = (unsigned)(uintptr_t)(&ring[tid]);

    // ---- Prime the HBM->L2 prefetch window (no counter cost). ----
#pragma unroll
    for (int p = 0; p < PF_DIST; ++p) {
        __builtin_prefetch((const void*)(xp + (size_t)p * S4), 0, 1);
    }

    // ---- Prime the async L2->LDS ring: steps 0..DEPTH-1 -> slots 0..23. ----
#pragma unroll
    for (int s = 0; s < DEPTH; ++s) {
        ASYNC_LOAD_B128(lds_lane + (unsigned)s * (BLK * 16),
                        (const void*)(xp + (size_t)s * S4));
    }

    // ---- Main loop: steps 0 .. T-DEPTH-1 (refill keeps depth constant). ----
#pragma unroll 4
    for (int t = 0; t < T_STEPS - DEPTH; ++t) {
        // Oldest DEPTH-th async load (step t) is complete when cnt <= DEPTH-1.
        asm volatile("s_wait_asynccnt 23" ::: "memory");

        v4f xt = ring[(t & (RING - 1)) * BLK + tid];   // ds_load_b128

        v4f a;
        a.x = __builtin_fmaf(wv.x, hv.x, xt.x);
        a.y = __builtin_fmaf(wv.y, hv.y, xt.y);
        a.z = __builtin_fmaf(wv.z, hv.z, xt.z);
        a.w = __builtin_fmaf(wv.w, hv.w, xt.w);
        hv.x = __builtin_fmaxf(a.x, 0.0f);
        hv.y = __builtin_fmaxf(a.y, 0.0f);
        hv.z = __builtin_fmaxf(a.z, 0.0f);
        hv.w = __builtin_fmaxf(a.w, 0.0f);

        __builtin_nontemporal_store(hv, op + (size_t)t * S4);

        // Prefetch far ahead (clamped in-bounds; lines just re-touch L2 at end).
        int pstep = t + PF_DIST;
        pstep = (pstep < T_STEPS) ? pstep : (T_STEPS - 1);
        __builtin_prefetch((const void*)(xp + (size_t)pstep * S4), 0, 1);

        // Refill: step t+DEPTH -> slot (t+DEPTH)&31 (consumed 8 steps ago).
        const int rstep = t + DEPTH;
        ASYNC_LOAD_B128(lds_lane + (unsigned)(rstep & (RING - 1)) * (BLK * 16),
                        (const void*)(xp + (size_t)rstep * S4));
    }

    // ---- Tail: drain all outstanding asyncs once, then consume the ring. ----
    asm volatile("s_wait_asynccnt 0" ::: "memory");
#pragma unroll
    for (int t = T_STEPS - DEPTH; t < T_STEPS; ++t) {
        v4f xt = ring[(t & (RING - 1)) * BLK + tid];

        v4f a;
        a.x = __builtin_fmaf(wv.x, hv.x, xt.x);
        a.y = __builtin_fmaf(wv.y, hv.y, xt.y);
        a.z = __builtin_fmaf(wv.z, hv.z, xt.z);
        a.w = __builtin_fmaf(wv.w, hv.w, xt.w);
        hv.x = __builtin_fmaxf(a.x, 0.0f);
        hv.y = __builtin_fmaxf(a.y, 0.0f);
        hv.z = __builtin_fmaxf(a.z, 0.0f);
        hv.w = __builtin_fmaxf(a.w, 0.0f);

        __builtin_nontemporal_store(hv, op + (size_t)t * S4);
    }
}

extern "C" void kernel_launch(void* const* d_in, const int* in_sizes, int n_in,
                              void* d_out, int out_size, void* d_ws, size_t ws_size,
                              hipStream_t stream) {
    (void)in_sizes; (void)n_in; (void)out_size; (void)d_ws; (void)ws_size;

    const float* x  = (const float*)d_in[0];  // [T, B, H]
    const float* h0 = (const float*)d_in[1];  // [B, H]
    const float* w  = (const float*)d_in[2];  // [H]
    float* out      = (float*)d_out;          // [T, B, H]

    // 16384 float4 columns -> 256 workgroups of 64 threads, spread wide
    // across WGPs for maximum memory-level parallelism.
    dim3 grid(S4 / BLK);
    dim3 block(BLK);
    indrnn_async_kernel<<<grid, block, 0, stream>>>(x, h0, w, out);
}